// GCN_3530463118085
// MI455X (gfx1250) — compile-verified
//
#include <hip/hip_runtime.h>
#include <hip/hip_bf16.h>

typedef __attribute__((ext_vector_type(2))) float v2f;
typedef __attribute__((ext_vector_type(8))) float v8f;

#define N_NODES  100000
#define N_EDGES  1600000
#define N_FEAT   32
#define HID      32
#define N_GRAPHS 1024

// ---------------------------------------------------------------------------
// Degree / normalization
// ---------------------------------------------------------------------------
__global__ void k_init_deg(float* __restrict__ deg, int n) {
    int i = blockIdx.x * blockDim.x + threadIdx.x;
    if (i < n) deg[i] = 1.0f;                     // self-loop counts once
}

__global__ void k_deg_accum(const int* __restrict__ dst, float* __restrict__ deg, int ne) {
    int e = blockIdx.x * blockDim.x + threadIdx.x;
    if (e < ne) unsafeAtomicAdd(&deg[dst[e]], 1.0f);
}

__global__ void k_deg_rsqrt(float* __restrict__ deg, int n) {
    int i = blockIdx.x * blockDim.x + threadIdx.x;
    if (i < n) deg[i] = __frsqrt_rn(deg[i]);      // deg >= 1 always
}

// ---------------------------------------------------------------------------
// Node transform: H[n,0:32] = (relu?)X[n,0:32] @ W[32,32]
// FP32 WMMA 16x16x4. One wave -> 16 rows x 32 cols.
// ISA 32-bit A 16x4 layout: lanes 0-15 hold K={0,1}, lanes 16-31 K={2,3}
// (vgpr0 = K even, vgpr1 = K odd). B 4x16 mirrors with N = lane&15.
// C/D 16x16: n = lane&15, m = vgpr + 8*(lane>>4).
// ---------------------------------------------------------------------------
template <bool RELU>
__global__ __launch_bounds__(256)
void k_transform(const float* __restrict__ X, const float* __restrict__ W,
                 float* __restrict__ H, int n_nodes) {
    const int lane = threadIdx.x & 31;
    const int wave = threadIdx.x >> 5;
    const int tile = blockIdx.x * 8 + wave;
    const int ntiles = (n_nodes + 15) >> 4;
    if (tile >= ntiles) return;                   // wave-uniform: EXEC stays all-1s for WMMA
    const int lo = lane & 15;
    const int hi = lane >> 4;                     // 0: K base +0, 1: K base +2
    const int rowbase = tile * 16;

    const float* xrow = X + (size_t)(rowbase + lo) * N_FEAT;
    v8f c0 = {}; v8f c1 = {};

#pragma unroll
    for (int kk = 0; kk < 8; ++kk) {
        const int k0 = kk * 4 + 2 * hi;
        float ax = xrow[k0], ay = xrow[k0 + 1];
        if (RELU) { ax = fmaxf(ax, 0.0f); ay = fmaxf(ay, 0.0f); }
        v2f a; a.x = ax; a.y = ay;

        v2f b0, b1;
        b0.x = W[(k0    ) * HID + lo];
        b0.y = W[(k0 + 1) * HID + lo];
        b1.x = W[(k0    ) * HID + 16 + lo];
        b1.y = W[(k0 + 1) * HID + 16 + lo];

        c0 = __builtin_amdgcn_wmma_f32_16x16x4_f32(false, a, false, b0, (short)0, c0, false, false);
        c1 = __builtin_amdgcn_wmma_f32_16x16x4_f32(false, a, false, b1, (short)0, c1, false, false);
    }

#pragma unroll
    for (int i = 0; i < 8; ++i) {
        const int m = i + 8 * hi;
        H[(size_t)(rowbase + m) * HID + lo]       = c0[i];
        H[(size_t)(rowbase + m) * HID + 16 + lo]  = c1[i];
    }
}

// ---------------------------------------------------------------------------
// agg[n,f] = H[n,f] * dinv[n]^2 + b[f]   (self-loop message + bias)
// ---------------------------------------------------------------------------
__global__ void k_init_agg(const float* __restrict__ H, const float* __restrict__ dinv,
                           const float* __restrict__ b, float* __restrict__ agg, int n_nodes) {
    int idx = blockIdx.x * blockDim.x + threadIdx.x;
    if (idx >= n_nodes * HID) return;
    int n = idx >> 5, f = idx & 31;
    float di = dinv[n];
    agg[idx] = H[idx] * di * di + b[f];
}

// ---------------------------------------------------------------------------
// Edge scatter: one wave32 per edge, lane == feature (HID == 32).
// agg[dst,f] += H[src,f] * dinv[src]*dinv[dst]
// ---------------------------------------------------------------------------
__global__ __launch_bounds__(256)
void k_scatter(const float* __restrict__ H, const int* __restrict__ src,
               const int* __restrict__ dst, const float* __restrict__ dinv,
               float* __restrict__ agg, int ne) {
    int e = (blockIdx.x * blockDim.x + threadIdx.x) >> 5;
    int lane = threadIdx.x & 31;
    if (e >= ne) return;
    int s = src[e], d = dst[e];
    float norm = dinv[s] * dinv[d];
    float v = H[(size_t)s * HID + lane] * norm;
    unsafeAtomicAdd(&agg[(size_t)d * HID + lane], v);
}

// ---------------------------------------------------------------------------
// Pooling: pooled[batch[n],f] += relu(agg2[n,f])
// ---------------------------------------------------------------------------
__global__ void k_zero(float* __restrict__ p, int n) {
    int i = blockIdx.x * blockDim.x + threadIdx.x;
    if (i < n) p[i] = 0.0f;
}

__global__ void k_pool(const float* __restrict__ H, const int* __restrict__ batch,
                       float* __restrict__ pooled, int n_nodes) {
    int idx = blockIdx.x * blockDim.x + threadIdx.x;
    if (idx >= n_nodes * HID) return;
    int n = idx >> 5, f = idx & 31;
    float v = fmaxf(H[idx], 0.0f);
    unsafeAtomicAdd(&pooled[(size_t)batch[n] * HID + f], v);
}

// ---------------------------------------------------------------------------
// Readout: out[g] = pooled[g,:] . Wout + bout
// ---------------------------------------------------------------------------
__global__ void k_readout(const float* __restrict__ pooled, const float* __restrict__ Wout,
                          const float* __restrict__ bout, float* __restrict__ out, int ng) {
    int g = blockIdx.x * blockDim.x + threadIdx.x;
    if (g >= ng) return;
    float acc = bout[0];
#pragma unroll
    for (int f = 0; f < HID; ++f) acc = fmaf(pooled[(size_t)g * HID + f], Wout[f], acc);
    out[g] = acc;
}

// ---------------------------------------------------------------------------
// Launch
// ---------------------------------------------------------------------------
extern "C" void kernel_launch(void* const* d_in, const int* in_sizes, int n_in,
                              void* d_out, int out_size, void* d_ws, size_t ws_size,
                              hipStream_t stream) {
    const float* x     = (const float*)d_in[0];
    const int*   eidx  = (const int*)  d_in[1];
    const int*   batch = (const int*)  d_in[2];
    const float* W1    = (const float*)d_in[3];
    const float* b1    = (const float*)d_in[4];
    const float* W2    = (const float*)d_in[5];
    const float* b2    = (const float*)d_in[6];
    const float* Wout  = (const float*)d_in[7];
    const float* bout  = (const float*)d_in[8];
    float* out = (float*)d_out;

    const int*  src = eidx;
    const int*  dst = eidx + N_EDGES;

    // workspace layout (floats)
    float* ws     = (float*)d_ws;
    float* dinv   = ws;                                   // 100000
    float* bufA   = ws + (1 << 17);                       // 100000*32 (transform out)
    float* bufB   = bufA + (size_t)N_NODES * HID;         // 100000*32 (aggregation)
    float* pooled = bufB + (size_t)N_NODES * HID;         // 1024*32

    const int T = 256;
    const int g_nodes  = (N_NODES + T - 1) / T;
    const int g_edges  = (N_EDGES + T - 1) / T;
    const int g_nodeF  = (N_NODES * HID + T - 1) / T;
    const int g_edgeW  = (N_EDGES * 32 + T - 1) / T;      // one wave per edge
    const int g_tiles  = (((N_NODES + 15) / 16) + 7) / 8; // 8 waves/block

    // normalization (shared by both convs)
    k_init_deg <<<g_nodes, T, 0, stream>>>(dinv, N_NODES);
    k_deg_accum<<<g_edges, T, 0, stream>>>(dst, dinv, N_EDGES);
    k_deg_rsqrt<<<g_nodes, T, 0, stream>>>(dinv, N_NODES);

    // layer 1
    k_transform<false><<<g_tiles, T, 0, stream>>>(x, W1, bufA, N_NODES);
    k_init_agg <<<g_nodeF, T, 0, stream>>>(bufA, dinv, b1, bufB, N_NODES);
    k_scatter  <<<g_edgeW, T, 0, stream>>>(bufA, src, dst, dinv, bufB, N_EDGES);

    // layer 2 (ReLU of layer-1 output fused into the WMMA A-load)
    k_transform<true><<<g_tiles, T, 0, stream>>>(bufB, W2, bufA, N_NODES);
    k_init_agg <<<g_nodeF, T, 0, stream>>>(bufA, dinv, b2, bufB, N_NODES);
    k_scatter  <<<g_edgeW, T, 0, stream>>>(bufA, src, dst, dinv, bufB, N_EDGES);

    // pool (ReLU fused) + readout
    k_zero   <<<(N_GRAPHS * HID + T - 1) / T, T, 0, stream>>>(pooled, N_GRAPHS * HID);
    k_pool   <<<g_nodeF, T, 0, stream>>>(bufB, batch, pooled, N_NODES);
    k_readout<<<(N_GRAPHS + T - 1) / T, T, 0, stream>>>(pooled, Wout, bout, out, N_GRAPHS);
}